// Hydra_46428596470380
// MI455X (gfx1250) — compile-verified
//
#include <hip/hip_runtime.h>
#include <math.h>
#include <stdint.h>

typedef __attribute__((ext_vector_type(2))) float v2f;
typedef __attribute__((ext_vector_type(8))) float v8f;

#define B_DIM 8
#define S_LEN 4096
#define D_DIM 2048
#define E_DIM 6144          // 3*D
#define ROT   1024          // rot_dim = D/2
#define M_TOT (B_DIM * S_LEN)
#define LN_THETA 9.210340371976184f   // ln(10000)
#define SCHUNKS 32
#define LDSTR 36            // 144B row stride: 16B-aligned for B128 async, conflict-free frags
#define KTILES (D_DIM / 32)

__device__ __forceinline__ void async_b128(unsigned lds_off, const void* gptr) {
  // GLOBAL_LOAD_ASYNC_TO_LDS_B128: LDS dst addr in VGPR, 64-bit global addr, ASYNCcnt-tracked
  asm volatile("global_load_async_to_lds_b128 %0, %1, off"
               :: "v"(lds_off), "v"((uint64_t)(uintptr_t)gptr) : "memory");
}
__device__ __forceinline__ void wait_async0() {
  asm volatile("s_wait_asynccnt 0x0" ::: "memory");
}

// ---------------------------------------------------------------------------
// K1: QKV[row,e] = sum_d X[row,d]*W_in[e,d] + b_in[e], then RoPE on Q,K
//     first ROT features.  128x128 block tile, 8 waves x (64x32) via
//     V_WMMA_F32_16X16X4_F32, double-buffered async-to-LDS staging.
// grid (M/128, E/128), block 256
// ---------------------------------------------------------------------------
__global__ __launch_bounds__(256) void qkv_gemm_rope(
    const float* __restrict__ X, const float* __restrict__ Win,
    const float* __restrict__ bin, float* __restrict__ QKV)
{
  __shared__ __align__(16) float Xs[2][128][LDSTR];
  __shared__ __align__(16) float Ws[2][128][LDSTR];

  const int tid   = threadIdx.x;
  const int lane  = tid & 31;
  const int wave  = tid >> 5;
  const int l16   = lane & 15;
  const int khalf = (lane >> 4) << 1;     // lanes 0-15 -> K{0,1}, 16-31 -> K{2,3}
  const int rhalf = (lane >> 4) << 3;     // C/D layout: upper lanes hold M=r+8
  const int rowBase = blockIdx.x * 128;
  const int colBase = blockIdx.y * 128;
  const int mW = (wave >> 2) * 64;
  const int nW = (wave & 3) * 32;

  v8f acc[4][2];
  #pragma unroll
  for (int i = 0; i < 4; ++i)
    #pragma unroll
    for (int j = 0; j < 2; ++j) acc[i][j] = (v8f)0.0f;

  // Per-thread staging map: 1024 16B-chunks per 128x32 tile, 4 per thread.
  const int sr  = (tid + 0 * 256) >> 3;         // row for chunk j: (tid+j*256)/8
  const int sc4 = (tid & 7) << 2;               // column (same for all j)

  // ---- prologue: async-stage tile 0 into buffer 0 ----
  #pragma unroll
  for (int j = 0; j < 4; ++j) {
    const int r = sr + j * 32;
    async_b128((unsigned)(uintptr_t)&Xs[0][r][sc4],
               &X[(size_t)(rowBase + r) * D_DIM + sc4]);
    async_b128((unsigned)(uintptr_t)&Ws[0][r][sc4],
               &Win[(size_t)(colBase + r) * D_DIM + sc4]);
  }
  wait_async0();
  __syncthreads();

  for (int t = 0; t < KTILES; ++t) {
    const int cur = t & 1;
    // ---- issue async copies for tile t+1 into the other buffer ----
    if (t + 1 < KTILES) {
      const int k1 = (t + 1) * 32;
      #pragma unroll
      for (int j = 0; j < 4; ++j) {
        const int r = sr + j * 32;
        async_b128((unsigned)(uintptr_t)&Xs[cur ^ 1][r][sc4],
                   &X[(size_t)(rowBase + r) * D_DIM + k1 + sc4]);
        async_b128((unsigned)(uintptr_t)&Ws[cur ^ 1][r][sc4],
                   &Win[(size_t)(colBase + r) * D_DIM + k1 + sc4]);
      }
    }
    // ---- compute on current buffer: 8 k-steps x 8 WMMAs ----
    #pragma unroll
    for (int kk = 0; kk < 32; kk += 4) {
      v2f a[4], b[2];
      #pragma unroll
      for (int fm = 0; fm < 4; ++fm) {
        const int m = mW + fm * 16 + l16;
        a[fm].x = Xs[cur][m][kk + khalf];
        a[fm].y = Xs[cur][m][kk + khalf + 1];
      }
      #pragma unroll
      for (int fn = 0; fn < 2; ++fn) {
        const int n = nW + fn * 16 + l16;
        b[fn].x = Ws[cur][n][kk + khalf];
        b[fn].y = Ws[cur][n][kk + khalf + 1];
      }
      #pragma unroll
      for (int fm = 0; fm < 4; ++fm)
        #pragma unroll
        for (int fn = 0; fn < 2; ++fn)
          acc[fm][fn] = __builtin_amdgcn_wmma_f32_16x16x4_f32(
              false, a[fm], false, b[fn], (short)0, acc[fm][fn], false, false);
    }
    if (t + 1 < KTILES) wait_async0();
    __syncthreads();
  }

  // Epilogue: bias + RoPE (pair partner = adjacent lane) + store.
  #pragma unroll
  for (int fm = 0; fm < 4; ++fm) {
    #pragma unroll
    for (int fn = 0; fn < 2; ++fn) {
      const int e  = colBase + nW + fn * 16 + l16;
      const int el = e & (D_DIM - 1);
      const bool doRope = (e < 2 * D_DIM) && (el < ROT);
      const float bias = bin[e];
      const float invf = expf(-(float)(el & ~1) * (LN_THETA / (float)ROT));
      #pragma unroll
      for (int r = 0; r < 8; ++r) {
        const int row = rowBase + mW + fm * 16 + r + rhalf;
        float val = acc[fm][fn][r] + bias;
        const float partner = __shfl_xor(val, 1, 32);
        float sv, cv;
        sincosf((float)(row & (S_LEN - 1)) * invf, &sv, &cv);
        const float roped = (e & 1) ? (val * cv + partner * sv)
                                    : (val * cv - partner * sv);
        QKV[(size_t)row * E_DIM + e] = doRope ? roped : val;
      }
    }
  }
}

// ---------------------------------------------------------------------------
// K2a: per-row inverse norm of K segment.  Wave per row, 8 rows per block.
// ---------------------------------------------------------------------------
__global__ __launch_bounds__(256) void k_norm(
    const float* __restrict__ QKV, float* __restrict__ ninv)
{
  const int lane = threadIdx.x & 31;
  const int wave = threadIdx.x >> 5;
  const int row  = blockIdx.x * 8 + wave;
  const float* kr = QKV + (size_t)row * E_DIM + D_DIM;
  float s = 0.0f;
  #pragma unroll 8
  for (int j = 0; j < D_DIM / 32; ++j) {
    const float v = kr[lane + j * 32];
    s += v * v;
  }
  #pragma unroll
  for (int off = 16; off > 0; off >>= 1) s += __shfl_xor(s, off, 32);
  if (lane == 0) ninv[row] = 1.0f / fmaxf(sqrtf(s), 1e-12f);
}

// ---------------------------------------------------------------------------
// K2b: partial A over S chunks (deterministic, no atomics).
// grid (D/256, SCHUNKS, B), block 256, thread-per-feature-column
// ---------------------------------------------------------------------------
__global__ __launch_bounds__(256) void a_partial(
    const float* __restrict__ QKV, const float* __restrict__ ninv,
    float* __restrict__ part)
{
  const int d  = blockIdx.x * 256 + threadIdx.x;
  const int b  = blockIdx.z;
  const int s0 = blockIdx.y * (S_LEN / SCHUNKS);
  float acc = 0.0f;
  for (int s = 0; s < S_LEN / SCHUNKS; ++s) {
    const int row = b * S_LEN + s0 + s;
    const size_t roff = (size_t)row * E_DIM;
    acc += QKV[roff + D_DIM + d] * ninv[row] * QKV[roff + 2 * D_DIM + d];
  }
  part[(size_t)(b * SCHUNKS + blockIdx.y) * D_DIM + d] = acc;
}

// K2c: fold chunk partials -> A[b,d].  grid (B*D/256)
__global__ __launch_bounds__(256) void a_reduce(
    const float* __restrict__ part, float* __restrict__ A)
{
  const int idx = blockIdx.x * 256 + threadIdx.x;   // b*D + d
  const int b = idx >> 11, d = idx & (D_DIM - 1);
  float s = 0.0f;
  #pragma unroll 4
  for (int c = 0; c < SCHUNKS; ++c)
    s += part[(size_t)(b * SCHUNKS + c) * D_DIM + d];
  A[idx] = s;
}

// ---------------------------------------------------------------------------
// K3: out[row,e] = sum_d (A[b,d]*Q[row,d]) * W_out[e,d] + b_out[e] + X[row,e]
//     Same double-buffered async pipeline; A-scale applied to the A-fragment
//     from a small double-buffered LDS slice (all 128 rows share one b).
// grid (M/128, D/128), block 256
// ---------------------------------------------------------------------------
__global__ __launch_bounds__(256) void out_gemm(
    const float* __restrict__ QKV, const float* __restrict__ Avec,
    const float* __restrict__ Wout, const float* __restrict__ bout,
    const float* __restrict__ X, float* __restrict__ out)
{
  __shared__ __align__(16) float Xs[2][128][LDSTR];
  __shared__ __align__(16) float Ws[2][128][LDSTR];
  __shared__ float As[2][32];

  const int tid   = threadIdx.x;
  const int lane  = tid & 31;
  const int wave  = tid >> 5;
  const int l16   = lane & 15;
  const int khalf = (lane >> 4) << 1;
  const int rhalf = (lane >> 4) << 3;
  const int rowBase = blockIdx.x * 128;
  const int colBase = blockIdx.y * 128;
  const int mW = (wave >> 2) * 64;
  const int nW = (wave & 3) * 32;
  const int bOff = (rowBase >> 12) << 11;   // b * D

  v8f acc[4][2];
  #pragma unroll
  for (int i = 0; i < 4; ++i)
    #pragma unroll
    for (int j = 0; j < 2; ++j) acc[i][j] = (v8f)0.0f;

  const int sr  = tid >> 3;
  const int sc4 = (tid & 7) << 2;

  #pragma unroll
  for (int j = 0; j < 4; ++j) {
    const int r = sr + j * 32;
    async_b128((unsigned)(uintptr_t)&Xs[0][r][sc4],
               &QKV[(size_t)(rowBase + r) * E_DIM + sc4]);
    async_b128((unsigned)(uintptr_t)&Ws[0][r][sc4],
               &Wout[(size_t)(colBase + r) * D_DIM + sc4]);
  }
  if (tid < 32) As[0][tid] = Avec[bOff + tid];
  wait_async0();
  __syncthreads();

  for (int t = 0; t < KTILES; ++t) {
    const int cur = t & 1;
    if (t + 1 < KTILES) {
      const int k1 = (t + 1) * 32;
      #pragma unroll
      for (int j = 0; j < 4; ++j) {
        const int r = sr + j * 32;
        async_b128((unsigned)(uintptr_t)&Xs[cur ^ 1][r][sc4],
                   &QKV[(size_t)(rowBase + r) * E_DIM + k1 + sc4]);
        async_b128((unsigned)(uintptr_t)&Ws[cur ^ 1][r][sc4],
                   &Wout[(size_t)(colBase + r) * D_DIM + k1 + sc4]);
      }
      if (tid < 32) As[cur ^ 1][tid] = Avec[bOff + k1 + tid];
    }
    #pragma unroll
    for (int kk = 0; kk < 32; kk += 4) {
      const float sa0 = As[cur][kk + khalf];
      const float sa1 = As[cur][kk + khalf + 1];
      v2f a[4], b[2];
      #pragma unroll
      for (int fm = 0; fm < 4; ++fm) {
        const int m = mW + fm * 16 + l16;
        a[fm].x = Xs[cur][m][kk + khalf] * sa0;
        a[fm].y = Xs[cur][m][kk + khalf + 1] * sa1;
      }
      #pragma unroll
      for (int fn = 0; fn < 2; ++fn) {
        const int n = nW + fn * 16 + l16;
        b[fn].x = Ws[cur][n][kk + khalf];
        b[fn].y = Ws[cur][n][kk + khalf + 1];
      }
      #pragma unroll
      for (int fm = 0; fm < 4; ++fm)
        #pragma unroll
        for (int fn = 0; fn < 2; ++fn)
          acc[fm][fn] = __builtin_amdgcn_wmma_f32_16x16x4_f32(
              false, a[fm], false, b[fn], (short)0, acc[fm][fn], false, false);
    }
    if (t + 1 < KTILES) wait_async0();
    __syncthreads();
  }

  #pragma unroll
  for (int fm = 0; fm < 4; ++fm) {
    #pragma unroll
    for (int fn = 0; fn < 2; ++fn) {
      const int e = colBase + nW + fn * 16 + l16;
      const float bias = bout[e];
      #pragma unroll
      for (int r = 0; r < 8; ++r) {
        const int row = rowBase + mW + fm * 16 + r + rhalf;
        out[(size_t)row * D_DIM + e] =
            acc[fm][fn][r] + bias + X[(size_t)row * D_DIM + e];
      }
    }
  }
}

// ---------------------------------------------------------------------------
extern "C" void kernel_launch(void* const* d_in, const int* in_sizes, int n_in,
                              void* d_out, int out_size, void* d_ws, size_t ws_size,
                              hipStream_t stream) {
  const float* X    = (const float*)d_in[0];
  const float* Win  = (const float*)d_in[1];
  const float* bin  = (const float*)d_in[2];
  const float* Wout = (const float*)d_in[3];
  const float* bout = (const float*)d_in[4];
  float* out = (float*)d_out;

  // Workspace layout (floats)
  float* ws = (float*)d_ws;
  const size_t QKV_N = (size_t)M_TOT * E_DIM;          // 201,326,592
  float* qkv  = ws;
  float* ninv = ws + QKV_N;                            // M_TOT
  float* part = ninv + M_TOT;                          // B*SCHUNKS*D
  float* A    = part + (size_t)B_DIM * SCHUNKS * D_DIM;// B*D

  qkv_gemm_rope<<<dim3(M_TOT / 128, E_DIM / 128), 256, 0, stream>>>(
      X, Win, bin, qkv);
  k_norm<<<M_TOT / 8, 256, 0, stream>>>(qkv, ninv);
  a_partial<<<dim3(D_DIM / 256, SCHUNKS, B_DIM), 256, 0, stream>>>(
      qkv, ninv, part);
  a_reduce<<<(B_DIM * D_DIM) / 256, 256, 0, stream>>>(part, A);
  out_gemm<<<dim3(M_TOT / 128, D_DIM / 128), 256, 0, stream>>>(
      qkv, A, Wout, bout, X, out);
}